// TripletLoss_dist_69672959475957
// MI455X (gfx1250) — compile-verified
//
#include <hip/hip_runtime.h>
#include <hip/hip_bf16.h>
#include <math.h>
#include <stdint.h>

typedef __attribute__((ext_vector_type(2))) float v2f;
typedef __attribute__((ext_vector_type(8))) float v8f;

#define MARGIN_F 5.0f
#define NROWS 8192
#define DCOLS 4096
#define CHUNK 128                  // columns staged per double-buffer step
#define NCHUNK (DCOLS / CHUNK)     // 32
#define RPITCH 132                 // 128 + 4 pad dwords: 16-row ds_load_b64 stride = 4 mod 64 banks

// One contiguous 16B per lane, DMA'd straight into LDS (ASYNCcnt-tracked).
__device__ __forceinline__ void async_ld16(uint32_t lds_byte_off, const float* g)
{
    asm volatile("global_load_async_to_lds_b128 %0, %1, off"
                 :: "v"(lds_byte_off), "v"(g) : "memory");
}

// ---------------------------------------------------------------------------
// Phase 1: per-row squared-L2 on the matrix pipe, fed by async LDS staging.
// Block = 16 rows. Stage 16x128 f32 tiles of recon & x (coalesced B128 DMA),
// then 8 waves each run 4 WMMA steps per tile:
//   A[m][k] = (recon-x)^2 (from LDS), B = ones  =>  D[m][*] += row sums.
// A layout (32-bit 16x4): lanes 0-15: M=lane,K{0,1}; lanes 16-31: M=lane-16,K{2,3}.
// ---------------------------------------------------------------------------
__global__ __launch_bounds__(256) void rowdist_wmma_kernel(
    const float* __restrict__ recon, const float* __restrict__ x,
    float* __restrict__ dvec)
{
    __shared__ float rbuf[2][16 * RPITCH];
    __shared__ float xbuf[2][16 * RPITCH];
    __shared__ float part[8][16];

    const int tid   = threadIdx.x;
    const int wave  = tid >> 5;
    const int lane  = tid & 31;
    const int m     = lane & 15;   // WMMA row
    const int khalf = lane >> 4;   // 0 -> K{0,1}, 1 -> K{2,3}
    const int row_base = blockIdx.x * 16;

    // staging role: thread t copies row t>>4, 16B-segments (t&15) and (t&15)+16
    const int srow = tid >> 4;
    const int sseg = tid & 15;

    const uint32_t rlds0 = (uint32_t)(uintptr_t)(void*)&rbuf[0][0];
    const uint32_t xlds0 = (uint32_t)(uintptr_t)(void*)&xbuf[0][0];
    const size_t   grow  = (size_t)(row_base + srow) * DCOLS;

    v8f c = {};
    v2f bones; bones.x = 1.0f; bones.y = 1.0f;   // all-ones B => WMMA row sums

    // prologue: DMA chunk 0 into buffer 0 (4 async ops per wave)
    {
        const uint32_t lb = (uint32_t)(srow * (RPITCH * 4) + sseg * 16);
        const size_t   gb = grow + (size_t)sseg * 4;
        async_ld16(rlds0 + lb,       recon + gb);
        async_ld16(rlds0 + lb + 256, recon + gb + 64);
        async_ld16(xlds0 + lb,       x + gb);
        async_ld16(xlds0 + lb + 256, x + gb + 64);
    }

    for (int ch = 0; ch < NCHUNK; ++ch) {
        if (ch + 1 < NCHUNK) {
            // issue next tile's DMA, then wait for current tile only (<=4 newest outstanding)
            const uint32_t lb = (uint32_t)(((ch + 1) & 1) * (16 * RPITCH * 4)
                                           + srow * (RPITCH * 4) + sseg * 16);
            const size_t   gb = grow + (size_t)(ch + 1) * CHUNK + (size_t)sseg * 4;
            async_ld16(rlds0 + lb,       recon + gb);
            async_ld16(rlds0 + lb + 256, recon + gb + 64);
            async_ld16(xlds0 + lb,       x + gb);
            async_ld16(xlds0 + lb + 256, x + gb + 64);
            asm volatile("s_wait_asynccnt 0x4" ::: "memory");
        } else {
            asm volatile("s_wait_asynccnt 0x0" ::: "memory");
        }
        __syncthreads();

        const int buf = ch & 1;
        const int kw  = wave * 16 + khalf * 2;
        #pragma unroll
        for (int s = 0; s < 4; ++s) {
            const int k = kw + s * 4;
            v2f rv = *(const v2f*)&rbuf[buf][m * RPITCH + k];  // ds_load_b64, bank-conflict-free
            v2f xv = *(const v2f*)&xbuf[buf][m * RPITCH + k];
            v2f e  = rv - xv;
            v2f a  = e * e;
            c = __builtin_amdgcn_wmma_f32_16x16x4_f32(
                    /*neg_a=*/false, a, /*neg_b=*/false, bones,
                    /*c_mod=*/(short)0, c, /*reuse_a=*/false, /*reuse_b=*/false);
        }
        __syncthreads();   // protect buffer from next iteration's DMA
    }

    // D layout: VGPR r of lanes 0-15 holds M=r; lanes 16-31 hold M=r+8.
    if (lane == 0) {
        #pragma unroll
        for (int r = 0; r < 8; ++r) part[wave][r] = c[r];
    } else if (lane == 16) {
        #pragma unroll
        for (int r = 0; r < 8; ++r) part[wave][8 + r] = c[r];
    }
    __syncthreads();

    if (tid < 16) {
        float s = 0.0f;
        #pragma unroll
        for (int w = 0; w < 8; ++w) s += part[w][tid];
        dvec[row_base + tid] = sqrtf(s) + 0.001f;
    }
}

// ---------------------------------------------------------------------------
// Phase 2: all pos-neg pairs of max(0, d_pos - d_neg + margin).
// d[] is 32KB -> tile through LDS. f64 atomic accumulator for precision.
// ---------------------------------------------------------------------------
__global__ __launch_bounds__(256) void pairloss_kernel(
    const float* __restrict__ dvec, const int* __restrict__ tgt,
    double* __restrict__ acc, int* __restrict__ cnts)
{
    __shared__ float sd[256];
    __shared__ int   st[256];

    const int   i  = blockIdx.x * 256 + threadIdx.x;
    const float di = dvec[i];
    const int   ti = tgt[i];

    float local = 0.0f;
    for (int jt = 0; jt < NROWS; jt += 256) {
        sd[threadIdx.x] = dvec[jt + threadIdx.x];
        st[threadIdx.x] = tgt[jt + threadIdx.x];
        __syncthreads();
        if (ti == 1) {
            #pragma unroll 8
            for (int j = 0; j < 256; ++j) {
                float v = di - sd[j] + MARGIN_F;
                v = v > 0.0f ? v : 0.0f;
                local += (st[j] == 0) ? v : 0.0f;
            }
        }
        __syncthreads();
    }

    atomicAdd(acc, (double)local);          // GLOBAL_ATOMIC_ADD_F64
    if (ti == 1) atomicAdd(&cnts[0], 1);    // n_pos
    if (ti == 0) atomicAdd(&cnts[1], 1);    // n_neg
}

__global__ void init_kernel(double* acc, int* cnts)
{
    *acc = 0.0;
    cnts[0] = 0;
    cnts[1] = 0;
}

__global__ void finalize_kernel(const double* __restrict__ acc,
                                const int* __restrict__ cnts,
                                float* __restrict__ out)
{
    double cnt = (double)cnts[0] * (double)cnts[1];
    out[0] = (float)(acc[0] / cnt);
}

// ---------------------------------------------------------------------------
// Workspace layout: [0, 32KB) d[8192] f32 | acc f64 | cnts int[2]
// ---------------------------------------------------------------------------
extern "C" void kernel_launch(void* const* d_in, const int* in_sizes, int n_in,
                              void* d_out, int out_size, void* d_ws, size_t ws_size,
                              hipStream_t stream) {
    (void)in_sizes; (void)n_in; (void)out_size; (void)ws_size;

    const float* recon = (const float*)d_in[0];
    const float* x     = (const float*)d_in[1];
    const int*   tgt   = (const int*)d_in[2];

    float*  dvec = (float*)d_ws;
    double* acc  = (double*)((char*)d_ws + NROWS * sizeof(float)); // 8B aligned
    int*    cnts = (int*)(acc + 1);
    float*  out  = (float*)d_out;

    init_kernel<<<1, 1, 0, stream>>>(acc, cnts);
    rowdist_wmma_kernel<<<NROWS / 16, 256, 0, stream>>>(recon, x, dvec);
    pairloss_kernel<<<NROWS / 256, 256, 0, stream>>>(dvec, tgt, acc, cnts);
    finalize_kernel<<<1, 1, 0, stream>>>(acc, cnts, out);
}